// PINNLoss_21887153341067
// MI455X (gfx1250) — compile-verified
//
#include <hip/hip_runtime.h>

// ---------------------------------------------------------------------------
// PINN RC-thermal loss, MI455X (gfx1250, wave32).
// Linear scan T[t+1] = ALPHA*T[t] + u[t] evaluated 16 steps at a time as a
// 16x16 matmul against a constant lower-triangular matrix of ALPHA powers,
// using chained V_WMMA_F32_16X16X4_F32 (full f32 precision).
// Interior tiles (547 of 548) use a branch-free specialized body; only the
// final ragged tile pays bounds predication.
// ---------------------------------------------------------------------------

typedef __attribute__((ext_vector_type(2))) float v2f;
typedef __attribute__((ext_vector_type(8))) float v8f;

#define B_ROWS 2048
#define T_LEN  8760
#define NT     548                      // ceil(8760/16); last tile has 8 valid cols
#define NWAVE  8
#define ROWS_PER_BLOCK (NWAVE * 16)     // 128
#define NBLOCK (B_ROWS / ROWS_PER_BLOCK)// 16

// physics constants (match reference, f32)
#define C_VAL   6.0e8f                  // 300000*2000
#define INV_R   2.0f                    // 1/R, R = 0.5
#define A_COEF  1.2e-5f                 // DT/(C*R) = 3600/3e8
#define B_COEF  1.0e-11f                // DT*1000/C^2
#define ALPHA_C (1.0f - A_COEF)
#define Y_COEF  (1000.0f / C_VAL)       // 1000/C in rhs
#define RESID_K (1.0f / C_VAL - 1.0f)   // dT - rhs == rhs*(1/C - 1) analytically
#define T_SETP  21.1111111f
#define T0_VAL  22.0f

// per-wave LDS layout (floats)
#define Y_STRIDE  18                    // 16 cols + 2 halo
#define O_STRIDE  17
#define WSZ (16*Y_STRIDE + 16*O_STRIDE + 16*O_STRIDE + 16)   // 848

__device__ __forceinline__ float apow_i(int m) {  // ALPHA^m, m in [0,15]
    float r = 1.0f;
    for (int i = 0; i < m; ++i) r *= ALPHA_C;
    return r;
}

// One 16-row x 16-timestep tile. FULL=true: all indices statically in range
// (no predication, prefetch next tile). FULL=false: ragged tail tile.
template <bool FULL>
__device__ __forceinline__ void process_tile(
    const float* __restrict__ y, const float* __restrict__ yt,
    const float* __restrict__ to,
    int row0, int t0, int lane, int hi, int l,
    float* __restrict__ yT, float* __restrict__ toT,
    float* __restrict__ ytT, float* __restrict__ cry,
    const v2f& bA, const v2f (&bK)[4],
    float& Tin, float& acc_data, float& acc_rc, float& acc_cf, float& acc_sm)
{
    __builtin_amdgcn_wave_barrier();

    // ---- stage 16x16 tiles of y / y_true / T_out into LDS (coalesced) ----
    {
        const int r = lane >> 1, h = lane & 1;
        const size_t ro = (size_t)(row0 + r) * T_LEN;
        #pragma unroll
        for (int q = 0; q < 2; ++q) {
            const int c   = h * 8 + q * 4;
            const int col = t0 + c;
            float4 vy, vt, vr;
            if (FULL || (col + 3 < T_LEN)) {
                vy = *reinterpret_cast<const float4*>(y  + ro + col);
                vr = *reinterpret_cast<const float4*>(yt + ro + col);
                vt = *reinterpret_cast<const float4*>(to + ro + col);
                if (FULL) {   // speculative prefetch of next tile's lines
                    __builtin_prefetch(y  + ro + col + 16, 0, 0);
                    __builtin_prefetch(yt + ro + col + 16, 0, 0);
                    __builtin_prefetch(to + ro + col + 16, 0, 0);
                }
            } else {
                vy.x = (col     < T_LEN) ? y [ro + col    ] : 0.0f;
                vy.y = (col + 1 < T_LEN) ? y [ro + col + 1] : 0.0f;
                vy.z = (col + 2 < T_LEN) ? y [ro + col + 2] : 0.0f;
                vy.w = (col + 3 < T_LEN) ? y [ro + col + 3] : 0.0f;
                vr.x = (col     < T_LEN) ? yt[ro + col    ] : 0.0f;
                vr.y = (col + 1 < T_LEN) ? yt[ro + col + 1] : 0.0f;
                vr.z = (col + 2 < T_LEN) ? yt[ro + col + 2] : 0.0f;
                vr.w = (col + 3 < T_LEN) ? yt[ro + col + 3] : 0.0f;
                vt.x = (col     < T_LEN) ? to[ro + col    ] : 0.0f;
                vt.y = (col + 1 < T_LEN) ? to[ro + col + 1] : 0.0f;
                vt.z = (col + 2 < T_LEN) ? to[ro + col + 2] : 0.0f;
                vt.w = (col + 3 < T_LEN) ? to[ro + col + 3] : 0.0f;
            }
            yT [r * Y_STRIDE + c] = vy.x; yT [r * Y_STRIDE + c + 1] = vy.y;
            yT [r * Y_STRIDE + c + 2] = vy.z; yT [r * Y_STRIDE + c + 3] = vy.w;
            ytT[r * O_STRIDE + c] = vr.x; ytT[r * O_STRIDE + c + 1] = vr.y;
            ytT[r * O_STRIDE + c + 2] = vr.z; ytT[r * O_STRIDE + c + 3] = vr.w;
            toT[r * O_STRIDE + c] = vt.x; toT[r * O_STRIDE + c + 1] = vt.y;
            toT[r * O_STRIDE + c + 2] = vt.z; toT[r * O_STRIDE + c + 3] = vt.w;
        }
        // y halo (cols 16,17) for the second difference
        if (lane < 16) {
            const size_t rh = (size_t)(row0 + lane) * T_LEN;
            #pragma unroll
            for (int e = 0; e < 2; ++e) {
                const int col = t0 + 16 + e;
                yT[lane * Y_STRIDE + 16 + e] =
                    (FULL || col < T_LEN) ? y[rh + col] : 0.0f;
            }
        }
    }
    __builtin_amdgcn_wave_barrier();
    asm volatile("s_wait_dscnt 0" ::: "memory");

    // ---- WMMA scan: T_tile = Tin * alpha_row + U * W  (5 x 16x16x4 f32) ----
    v2f aT;
    aT.x = (hi == 0) ? Tin : 0.0f;   // A column K=0 = carry-in, rest 0
    aT.y = 0.0f;
    v8f acc = {0.f, 0.f, 0.f, 0.f, 0.f, 0.f, 0.f, 0.f};
    acc = __builtin_amdgcn_wmma_f32_16x16x4_f32(
              false, aT, false, bA, (short)0, acc, false, false);
    #pragma unroll
    for (int k = 0; k < 4; ++k) {
        const int ci = 4 * k + 2 * hi;
        v2f aK;
        aK.x = A_COEF * toT[l * O_STRIDE + ci    ] + B_COEF * yT[l * Y_STRIDE + ci    ];
        aK.y = A_COEF * toT[l * O_STRIDE + ci + 1] + B_COEF * yT[l * Y_STRIDE + ci + 1];
        acc = __builtin_amdgcn_wmma_f32_16x16x4_f32(
                  false, aK, false, bK[k], (short)0, acc, false, false);
    }
    // D layout: lane (l,hi), vgpr v  <->  row b = v + 8*hi, time col j = l

    // ---- loss accumulation in D layout ----
    const int t = t0 + l;
    #pragma unroll
    for (int v = 0; v < 8; ++v) {
        const int b = v + 8 * hi;
        const float Th  = acc[v];
        const float yv  = yT [b * Y_STRIDE + l];
        const float y1  = yT [b * Y_STRIDE + l + 1];
        const float y2  = yT [b * Y_STRIDE + l + 2];
        const float tov = toT[b * O_STRIDE + l];
        const float ytv = ytT[b * O_STRIDE + l];
        if (FULL || t < T_LEN) {
            const float dd = yv - ytv;
            acc_data += dd * dd;
            const float cf = fabsf(Th - T_SETP) - 2.0f;
            acc_cf += fmaxf(cf, 0.0f);
        }
        if (FULL || t < T_LEN - 1) {
            const float rhs = (tov - Th) * INV_R + Y_COEF * yv;
            const float rs  = rhs * RESID_K;      // == dT - rhs analytically
            acc_rc += rs * rs;
        }
        if (FULL || t < T_LEN - 2) {
            const float d2 = y2 - 2.0f * y1 + yv;
            acc_sm += d2 * d2;
        }
    }

    // ---- carry to next tile: Tin' = ALPHA*T_hat[t0+15] + u[t0+15] ----
    if (FULL) {   // no carry needed after the final tile
        if (l == 15) {
            #pragma unroll
            for (int v = 0; v < 8; ++v) cry[(hi << 3) + v] = acc[v];
        }
        __builtin_amdgcn_wave_barrier();
        asm volatile("s_wait_dscnt 0" ::: "memory");
        const float u15 = A_COEF * toT[l * O_STRIDE + 15]
                        + B_COEF * yT [l * Y_STRIDE + 15];
        Tin = ALPHA_C * cry[l] + u15;
    }
}

__global__ __launch_bounds__(256) void pinn_main_kernel(
    const float* __restrict__ y,      // y_hat   (B,T)
    const float* __restrict__ yt,     // y_true  (B,T)
    const float* __restrict__ to,     // T_out   (B,T)
    double* __restrict__ partial)     // NBLOCK*4 doubles
{
    __shared__ float  smem[NWAVE * WSZ];
    __shared__ double waveAcc[NWAVE * 4];

    const int tid  = threadIdx.x;
    const int w    = tid >> 5;
    const int lane = tid & 31;
    const int hi   = lane >> 4;       // half-wave (0/1)
    const int l    = lane & 15;       // 0..15

    float* yT  = smem + w * WSZ;      // 16 x 18 (with 2-col halo)
    float* toT = yT  + 16 * Y_STRIDE; // 16 x 17
    float* ytT = toT + 16 * O_STRIDE; // 16 x 17
    float* cry = ytT + 16 * O_STRIDE; // 16 carry values

    const int row0 = blockIdx.x * ROWS_PER_BLOCK + w * 16;

    // --- constant B fragments (4x16 f32 B: vgpr0 -> row K=2*hi, vgpr1 ->
    //     K=2*hi+1, column N = lane%16; mirrors documented 16x4 A layout) ---
    // B_alpha[K][j] = (K==0) ? ALPHA^j : 0          (homogeneous carry-in term)
    // B_k[K][j]     = (4k+K < j) ? ALPHA^(j-1-(4k+K)) : 0
    v2f bA;
    bA.x = (hi == 0) ? apow_i(l) : 0.0f;
    bA.y = 0.0f;
    v2f bK[4];
    #pragma unroll
    for (int k = 0; k < 4; ++k) {
        const int i0 = 4 * k + 2 * hi;
        bK[k].x = (i0     < l) ? apow_i(l - 1 - i0) : 0.0f;
        bK[k].y = (i0 + 1 < l) ? apow_i(l - 2 - i0) : 0.0f;
    }

    float Tin = T0_VAL;               // lane l carries T_hat[row0+l][t0]
    float acc_data = 0.0f, acc_rc = 0.0f, acc_cf = 0.0f, acc_sm = 0.0f;

    // 547 branch-free interior tiles + 1 ragged tail tile
    for (int tile = 0; tile < NT - 1; ++tile) {
        process_tile<true>(y, yt, to, row0, tile * 16, lane, hi, l,
                           yT, toT, ytT, cry, bA, bK,
                           Tin, acc_data, acc_rc, acc_cf, acc_sm);
    }
    process_tile<false>(y, yt, to, row0, (NT - 1) * 16, lane, hi, l,
                        yT, toT, ytT, cry, bA, bK,
                        Tin, acc_data, acc_rc, acc_cf, acc_sm);

    // ---- deterministic reduction: shfl butterfly -> wave -> block ----
    #pragma unroll
    for (int off = 16; off > 0; off >>= 1) {
        acc_data += __shfl_xor(acc_data, off, 32);
        acc_rc   += __shfl_xor(acc_rc,   off, 32);
        acc_cf   += __shfl_xor(acc_cf,   off, 32);
        acc_sm   += __shfl_xor(acc_sm,   off, 32);
    }
    if (lane == 0) {
        waveAcc[w * 4 + 0] = (double)acc_data;
        waveAcc[w * 4 + 1] = (double)acc_rc;
        waveAcc[w * 4 + 2] = (double)acc_cf;
        waveAcc[w * 4 + 3] = (double)acc_sm;
    }
    __syncthreads();
    if (tid == 0) {
        double s0 = 0.0, s1 = 0.0, s2 = 0.0, s3 = 0.0;
        for (int i = 0; i < NWAVE; ++i) {
            s0 += waveAcc[i * 4 + 0];
            s1 += waveAcc[i * 4 + 1];
            s2 += waveAcc[i * 4 + 2];
            s3 += waveAcc[i * 4 + 3];
        }
        partial[blockIdx.x * 4 + 0] = s0;
        partial[blockIdx.x * 4 + 1] = s1;
        partial[blockIdx.x * 4 + 2] = s2;
        partial[blockIdx.x * 4 + 3] = s3;
    }
}

__global__ void pinn_final_kernel(const double* __restrict__ partial,
                                  float* __restrict__ out)
{
    if (threadIdx.x == 0 && blockIdx.x == 0) {
        double sd = 0.0, sr = 0.0, sc = 0.0, ss = 0.0;
        for (int i = 0; i < NBLOCK; ++i) {     // fixed order -> deterministic
            sd += partial[i * 4 + 0];
            sr += partial[i * 4 + 1];
            sc += partial[i * 4 + 2];
            ss += partial[i * 4 + 3];
        }
        const double NBT = (double)B_ROWS * (double)T_LEN;
        const double l_data    = sd / NBT;
        const double l_rc      = sr / ((double)B_ROWS * (double)(T_LEN - 1));
        const double l_comfort = sc / NBT;
        const double l_smooth  = ss / ((double)B_ROWS * (double)(T_LEN - 2));
        out[0] = (float)(l_data + 1.0 * l_rc + 0.1 * l_comfort + 0.01 * l_smooth);
    }
}

extern "C" void kernel_launch(void* const* d_in, const int* in_sizes, int n_in,
                              void* d_out, int out_size, void* d_ws, size_t ws_size,
                              hipStream_t stream) {
    (void)in_sizes; (void)n_in; (void)out_size; (void)ws_size;
    const float* y_hat  = (const float*)d_in[0];
    const float* y_true = (const float*)d_in[1];
    const float* T_out  = (const float*)d_in[2];
    double* partial = (double*)d_ws;             // NBLOCK*4 doubles = 512 B
    float*  out     = (float*)d_out;

    pinn_main_kernel<<<NBLOCK, 256, 0, stream>>>(y_hat, y_true, T_out, partial);
    pinn_final_kernel<<<1, 32, 0, stream>>>(partial, out);
}